// CRFLayer_19258633355316
// MI455X (gfx1250) — compile-verified
//
#include <hip/hip_runtime.h>
#include <hip/hip_bf16.h>

typedef float v2f __attribute__((ext_vector_type(2)));
typedef float v8f __attribute__((ext_vector_type(8)));

#define TAGS  96
#define SEQ   2048
#define BATCH 128
#define ROWS  16      // batch rows per workgroup (M of WMMA)
#define NWAVE 6       // one wave per 16-wide N tile (6*16 = 96)

// ---------------------------------------------------------------------------
// Forward algorithm (log partition) via exp-domain WMMA GEMM.
// alpha'(j) = m + log( exp(alpha-m) . exp(T) )(j) + emit(j),  masked blend.
// ---------------------------------------------------------------------------
__global__ __launch_bounds__(NWAVE * 32)
void crf_forward_kernel(const float* __restrict__ emissions,
                        const float* __restrict__ mask,
                        const float* __restrict__ transitions,
                        const float* __restrict__ start_t,
                        const float* __restrict__ end_t,
                        float* __restrict__ log_den)
{
    __shared__ float alpha[ROWS][TAGS];          // running log-alphas
    __shared__ float embuf[2][ROWS][TAGS];       // double-buffered emission tiles
    __shared__ float rmax[ROWS];                 // per-row max for rescaling

    const int tid  = threadIdx.x;
    const int wave = tid >> 5;                   // 0..5
    const int lane = tid & 31;
    const int b0   = blockIdx.x * ROWS;
    const int n0   = wave * 16;                  // this wave's tag-column tile

    // ---- B matrix: exp(transitions), resident in VGPRs for the whole scan.
    // 16x16x4 f32 B tile layout: VGPR0 = {lanes0-15:K0, lanes16-31:K2},
    //                            VGPR1 = {lanes0-15:K1, lanes16-31:K3}
    const int bcol = n0 + (lane & 15);
    const int kb   = (lane < 16) ? 0 : 2;
    v2f Bt[24];
#pragma unroll
    for (int kk = 0; kk < 24; ++kk) {
        int k = kk * 4 + kb;
        Bt[kk].x = __expf(transitions[k * TAGS + bcol]);
        Bt[kk].y = __expf(transitions[(k + 1) * TAGS + bcol]);
    }

    // ---- async emission-tile fetch: ROWS*TAGS floats = 6144 B = 384 x b128,
    // 2 chunks per thread (192 threads).
    auto issue_tile = [&](int t, int buf) {
#pragma unroll
        for (int part = 0; part < 2; ++part) {
            int c = part * (NWAVE * 32) + tid;          // 16-byte chunk id
            int f = c * 4;                              // float index in tile
            int r = f / TAGS, j = f % TAGS;             // TAGS % 4 == 0: no straddle
            const float* gsrc = emissions + ((long)(b0 + r) * SEQ + t) * TAGS + j;
            unsigned ldst = (unsigned)(size_t)(&embuf[buf][r][j]);
            asm volatile("global_load_async_to_lds_b128 %0, %1, off"
                         :: "v"(ldst), "v"((unsigned long long)(size_t)gsrc)
                         : "memory");
        }
    };

    // ---- t = 0: alpha = start + emissions[:, 0, :]
    for (int i = tid; i < ROWS * TAGS; i += NWAVE * 32) {
        int r = i / TAGS, j = i % TAGS;
        alpha[r][j] = start_t[j] + emissions[((long)(b0 + r) * SEQ) * TAGS + j];
    }
    issue_tile(1, 0);
    __syncthreads();

    const int r    = lane & 15;                  // A-matrix row owned by this lane
    const int c0   = (lane < 16) ? 0 : 48;       // half-row scan base for reductions
    const int gcol = n0 + (lane & 15);           // C-tile column (global tag index)
    const int rb   = (lane < 16) ? 0 : 8;        // C-tile row base

    for (int t = 1; t < SEQ; ++t) {
        const int buf = (t - 1) & 1;
        if (t + 1 < SEQ) issue_tile(t + 1, t & 1);

        // per-row max (each wave redundantly; lane<->row matches A layout)
        float m = alpha[r][c0];
#pragma unroll 8
        for (int j = 1; j < 48; ++j) m = fmaxf(m, alpha[r][c0 + j]);
        m = fmaxf(m, __shfl_xor(m, 16, 32));
        if (wave == 0 && lane < 16) rmax[r] = m;

        // A build (exp rescale) fused with 24 chained f32 WMMAs
        v8f C = {};
#pragma unroll
        for (int kk = 0; kk < 24; ++kk) {
            int k = kk * 4 + kb;
            v2f A;
            A.x = __expf(alpha[r][k] - m);
            A.y = __expf(alpha[r][k + 1] - m);
            C = __builtin_amdgcn_wmma_f32_16x16x4_f32(
                    false, A, false, Bt[kk], (short)0, C, false, false);
        }

        // wait for step-t emission tile (t+1's 2 async ops may stay in flight)
        if (t + 1 < SEQ) asm volatile("s_wait_asynccnt 2" ::: "memory");
        else             asm volatile("s_wait_asynccnt 0" ::: "memory");
        __syncthreads();   // alpha reads done everywhere; tile + rmax visible

        // post-process C tile -> new alpha (masked blend)
#pragma unroll
        for (int rr = 0; rr < 8; ++rr) {
            int row  = rb + rr;
            float mk = mask[(long)(b0 + row) * SEQ + t];
            float nv = rmax[row] + __logf(C[rr]) + embuf[buf][row][gcol];
            float ov = alpha[row][gcol];
            alpha[row][gcol] = mk * nv + (1.0f - mk) * ov;
        }
        __syncthreads();
    }

    // ---- log_den[b] = logsumexp_j(alpha[b][j] + end[j])
    {
        float m2 = -3.4e38f;
#pragma unroll 8
        for (int j = 0; j < 48; ++j) m2 = fmaxf(m2, alpha[r][c0 + j] + end_t[c0 + j]);
        m2 = fmaxf(m2, __shfl_xor(m2, 16, 32));
        float s = 0.0f;
#pragma unroll 8
        for (int j = 0; j < 48; ++j) s += __expf(alpha[r][c0 + j] + end_t[c0 + j] - m2);
        s += __shfl_xor(s, 16, 32);
        if (wave == 0 && lane < 16) log_den[b0 + r] = m2 + __logf(s);
    }
}

// ---------------------------------------------------------------------------
// Gold-path score (numerator): pure gather + block reduction, one WG / batch.
// ---------------------------------------------------------------------------
__global__ __launch_bounds__(256)
void crf_score_kernel(const float* __restrict__ emissions,
                      const int*   __restrict__ tags,
                      const float* __restrict__ mask,
                      const float* __restrict__ transitions,
                      const float* __restrict__ start_t,
                      const float* __restrict__ end_t,
                      float* __restrict__ log_num)
{
    __shared__ float red[256];
    __shared__ float redm[256];
    const int b = blockIdx.x, tid = threadIdx.x;

    float acc = 0.0f, msum = 0.0f;
    for (int t = tid; t < SEQ; t += 256) {
        float mk = mask[(long)b * SEQ + t];
        msum += mk;
        if (t >= 1) {
            int tp = tags[(long)b * SEQ + t - 1];
            int tc = tags[(long)b * SEQ + t];
            acc += mk * (transitions[tp * TAGS + tc] +
                         emissions[((long)b * SEQ + t) * TAGS + tc]);
        }
    }
    red[tid] = acc; redm[tid] = msum;
    __syncthreads();
    for (int s = 128; s > 0; s >>= 1) {
        if (tid < s) { red[tid] += red[tid + s]; redm[tid] += redm[tid + s]; }
        __syncthreads();
    }
    if (tid == 0) {
        int t0 = tags[(long)b * SEQ];
        float sc = red[0] + start_t[t0] + emissions[(long)b * SEQ * TAGS + t0];
        int last = (int)(redm[0] + 0.5f) - 1;
        sc += end_t[tags[(long)b * SEQ + last]];
        log_num[b] = sc;
    }
}

// ---------------------------------------------------------------------------
// Final scalar: mean(log_den - log_num)
// ---------------------------------------------------------------------------
__global__ __launch_bounds__(BATCH)
void crf_final_kernel(const float* __restrict__ den,
                      const float* __restrict__ num,
                      float* __restrict__ out)
{
    __shared__ float red[BATCH];
    int tid = threadIdx.x;
    red[tid] = den[tid] - num[tid];
    __syncthreads();
    for (int s = BATCH / 2; s > 0; s >>= 1) {
        if (tid < s) red[tid] += red[tid + s];
        __syncthreads();
    }
    if (tid == 0) out[0] = red[0] / (float)BATCH;
}

extern "C" void kernel_launch(void* const* d_in, const int* in_sizes, int n_in,
                              void* d_out, int out_size, void* d_ws, size_t ws_size,
                              hipStream_t stream)
{
    const float* emissions   = (const float*)d_in[0];
    const int*   tags        = (const int*)  d_in[1];
    const float* mask        = (const float*)d_in[2];
    const float* transitions = (const float*)d_in[3];
    const float* start_t     = (const float*)d_in[4];
    const float* end_t       = (const float*)d_in[5];

    float* ws      = (float*)d_ws;
    float* log_den = ws;            // [BATCH]
    float* log_num = ws + BATCH;    // [BATCH]

    crf_forward_kernel<<<BATCH / ROWS, NWAVE * 32, 0, stream>>>(
        emissions, mask, transitions, start_t, end_t, log_den);
    crf_score_kernel<<<BATCH, 256, 0, stream>>>(
        emissions, tags, mask, transitions, start_t, end_t, log_num);
    crf_final_kernel<<<1, BATCH, 0, stream>>>(log_den, log_num, (float*)d_out);
}